// GFlowLoss_25726854103404
// MI455X (gfx1250) — compile-verified
//
#include <hip/hip_runtime.h>
#include <stdint.h>
#include <math.h>

// Problem constants (from reference)
#define NN     3072
#define NIN    16
#define NF     3056
#define NT     3056
#define APITCH 3072
#define ITERS  100
#define LRC    0.1f
#define BB1    0.9f
#define BB2    0.999f
#define AEPS   1e-8f
#define OEPS   0.1f
#define TOLC   1e-3f

#define NTILES 191         // NT / 16 column tiles
#define CS     4           // column segments (pass-1 grid.y)
#define TP     20          // LDS tile row pitch in floats (16B-aligned rows, bank-conflict-free)

typedef float v2f __attribute__((ext_vector_type(2)));
typedef float v8f __attribute__((ext_vector_type(8)));

__device__ __forceinline__ float fast_rcp(float x) { return __builtin_amdgcn_rcpf(x); }
__device__ __forceinline__ float sigmoidf(float x) { return fast_rcp(1.0f + __expf(-x)); }

__device__ __forceinline__ uint32_t lds_off32(const void* p) {
  // Low 32 bits of a flat shared-aperture address are the LDS byte offset.
  return (uint32_t)(uintptr_t)p;
}

// DPP16 row_shr reduction step: x += lane(x + n) within each 16-lane row, 0-fill OOB.
#define DPP_ROWSHR_ADD(x, ctrl)                                                     \
  ((x) + __int_as_float(__builtin_amdgcn_update_dpp(                                \
             0, __float_as_int(x), (ctrl), 0xF, 0xF, true)))

__device__ __forceinline__ float rowsum16(float x) {
  x = DPP_ROWSHR_ADD(x, 0x118);  // row_shr:8
  x = DPP_ROWSHR_ADD(x, 0x114);  // row_shr:4
  x = DPP_ROWSHR_ADD(x, 0x112);  // row_shr:2
  x = DPP_ROWSHR_ADD(x, 0x111);  // row_shr:1 -> lanes 0 and 16 hold their row totals
  return x;
}

// ---------------------------------------------------------------------------
// Init: copy GL/tau from inputs, zero Adam state + scalars
// ---------------------------------------------------------------------------
__global__ void k_init(const float* __restrict__ GL0, const float* __restrict__ tau0,
                       float* GL, float* mG, float* vG,
                       float* tau, float* mt, float* vt, float* scal) {
  size_t idx = (size_t)blockIdx.x * blockDim.x + threadIdx.x;
  size_t stride = (size_t)gridDim.x * blockDim.x;
  const size_t MAT = (size_t)NF * NT;
  for (size_t i = idx; i < MAT; i += stride) { GL[i] = GL0[i]; mG[i] = 0.f; vG[i] = 0.f; }
  if (idx < NN) { tau[idx] = tau0[idx]; mt[idx] = 0.f; vt[idx] = 0.f; }
  if (idx < 8) scal[idx] = 0.f;
}

// ---------------------------------------------------------------------------
// Per-iteration zeroing: Cg accumulator, l_order accumulator, done_prev <- done_cur
// ---------------------------------------------------------------------------
__global__ void k_zero(float* Cg, float* scal) {
  int k = blockIdx.x * blockDim.x + threadIdx.x;
  if (k < NT) Cg[k] = 0.f;
  if (k == 0) { scal[0] = scal[1]; scal[2] = 0.f; }
}

// ---------------------------------------------------------------------------
// Pass 1: row products (zero-tracked), WMMA row-sums of G*relu^2 and G*relu,
// DPP column-sums of G*relu. Tiles double-buffered via async global->LDS.
// Block = 256 threads (8 waves) = 16 rows x one column segment. Grid = 191 x 4.
// ---------------------------------------------------------------------------
struct __align__(16) KbShared {
  float tauv[NT];               // tau[NIN + j]
  float tauu[16];               // tau[r0 + m]
  float glt[8][2][16 * TP];     // per-wave double-buffered GL tiles
  float at [8][2][16 * TP];     // per-wave double-buffered A tiles
  float Pw [8][16];             // per-wave row-product partials
  float Zw [8][16];             // per-wave zero counts
  float R2w[8][16];             // per-wave rowsum(G*relu^2)
  float Rgw[8][16];             // per-wave rowsum(G*relu)
};

__global__ __launch_bounds__(256) void k_pass1(
    const float* __restrict__ A, const float* __restrict__ GL,
    const float* __restrict__ tau,
    float* Pp, float* Zp, float* Rp,   // per-segment partials [CS][NF]
    float* Cg, float* scal) {
  __shared__ KbShared s;
  const int tid  = threadIdx.x;
  const int wave = tid >> 5;
  const int lane = tid & 31;
  const int r0   = blockIdx.x * 16;
  const int seg  = blockIdx.y;

  for (int j = tid; j < NT; j += 256) s.tauv[j] = tau[NIN + j];
  if (tid < 16) s.tauu[tid] = tau[r0 + tid];
  __syncthreads();

  const int row16 = lane & 15;   // WMMA A-layout: M = lane%16
  const int hi    = lane >> 4;   // lane half selects K pair
  const float taui = s.tauu[row16];

  float prodAcc = 1.f, zcAcc = 0.f;
  v8f D2 = {0.f,0.f,0.f,0.f,0.f,0.f,0.f,0.f};
  v8f Dg = {0.f,0.f,0.f,0.f,0.f,0.f,0.f,0.f};
  const v2f ones = {1.f, 1.f};

  const uint32_t lgl[2] = { lds_off32(s.glt[wave][0]), lds_off32(s.glt[wave][1]) };
  const uint32_t la [2] = { lds_off32(s.at [wave][0]), lds_off32(s.at [wave][1]) };

  // Async global->LDS tile copy: 16 rows x 64B; 64 b128 segments, 2/lane/matrix.
  auto issue = [&](int tt, int b) {
    const int c0 = tt * 16;
#pragma unroll
    for (int k = 0; k < 2; k++) {
      int sidx = lane + 32 * k;
      int r = sidx >> 2, sg = sidx & 3;
      uint64_t ga = (uint64_t)(uintptr_t)(GL + (size_t)(r0 + r) * NT + c0 + sg * 4);
      uint32_t ld = lgl[b] + (uint32_t)(r * TP * 4 + sg * 16);
      asm volatile("global_load_async_to_lds_b128 %0, %1, off" :: "v"(ld), "v"(ga) : "memory");
      uint64_t gb = (uint64_t)(uintptr_t)(A + (size_t)(r0 + r) * APITCH + NIN + c0 + sg * 4);
      uint32_t lb = la[b] + (uint32_t)(r * TP * 4 + sg * 16);
      asm volatile("global_load_async_to_lds_b128 %0, %1, off" :: "v"(lb), "v"(gb) : "memory");
    }
  };

  const int tstart  = seg + CS * wave;   // in [0, 31], always < NTILES
  const int tstride = CS * 8;            // 32
  int buf = 0;
  issue(tstart, 0);

  for (int t = tstart; t < NTILES; t += tstride) {
    const int c0 = t * 16;
    const bool hasnext = (t + tstride) < NTILES;
    if (hasnext) {
      issue(t + tstride, buf ^ 1);
      asm volatile("s_wait_asynccnt 4" ::: "memory");   // current buffer resident
    } else {
      asm volatile("s_wait_asynccnt 0" ::: "memory");
    }
    const float* glt = s.glt[wave][buf];
    const float* at  = s.at [wave][buf];

#pragma unroll
    for (int q = 0; q < 4; q++) {
      const int c = 4 * q + 2 * hi;                       // K = 2*hi + {0,1}
      const float2 g2 = *(const float2*)&glt[row16 * TP + c];
      const float2 a2 = *(const float2*)&at [row16 * TP + c];
      const float g0 = sigmoidf(g2.x), g1 = sigmoidf(g2.y);
      const float f0 = 1.f - 2.f * a2.x * g0;
      const float f1 = 1.f - 2.f * a2.y * g1;
      zcAcc   += (f0 == 0.f) ? 1.f : 0.f;
      prodAcc *= (f0 == 0.f) ? 1.f : f0;
      zcAcc   += (f1 == 0.f) ? 1.f : 0.f;
      prodAcc *= (f1 == 0.f) ? 1.f : f1;
      float d0 = taui - s.tauv[c0 + c]     + OEPS; d0 = d0 > 0.f ? d0 : 0.f;
      float d1 = taui - s.tauv[c0 + c + 1] + OEPS; d1 = d1 > 0.f ? d1 : 0.f;
      v2f o2 = { g0 * d0 * d0, g1 * d1 * d1 };  // loss term
      v2f og = { g0 * d0,      g1 * d1      };  // tau-gradient term
      // Row-sum via 16x16x4 f32 WMMA against ones: D += o x 1 (accumulates over tiles)
      D2 = __builtin_amdgcn_wmma_f32_16x16x4_f32(false, o2, false, ones, (short)0, D2, false, false);
      Dg = __builtin_amdgcn_wmma_f32_16x16x4_f32(false, og, false, ones, (short)0, Dg, false, false);
      // Column sums of og: same column across each 16-lane half -> DPP row_shr tree
      float cx = rowsum16(og.x);
      float cy = rowsum16(og.y);
      if (row16 == 0) {
        atomicAdd(&Cg[c0 + c],     cx);
        atomicAdd(&Cg[c0 + c + 1], cy);
      }
    }
    buf ^= 1;
  }

  // Combine lane halves (cols split across halves, same row): multiply/add
  prodAcc *= __shfl_xor(prodAcc, 16, 32);
  zcAcc   += __shfl_xor(zcAcc,   16, 32);
  if (lane < 16) { s.Pw[wave][lane] = prodAcc; s.Zw[wave][lane] = zcAcc; }
  // D layout: lane 0 holds rows 0..7 in VGPRs 0..7; lane 16 holds rows 8..15
  if ((lane & 15) == 0) {
#pragma unroll
    for (int r = 0; r < 8; r++) {
      s.R2w[wave][hi * 8 + r] = D2[r];
      s.Rgw[wave][hi * 8 + r] = Dg[r];
    }
  }
  __syncthreads();

  if (tid < 16) {
    float P = 1.f, Z = 0.f, R2 = 0.f, RG = 0.f;
#pragma unroll
    for (int w = 0; w < 8; w++) {
      P *= s.Pw[w][tid]; Z += s.Zw[w][tid];
      R2 += s.R2w[w][tid]; RG += s.Rgw[w][tid];
    }
    Pp[(size_t)seg * NF + r0 + tid] = P;
    Zp[(size_t)seg * NF + r0 + tid] = Z;
    Rp[(size_t)seg * NF + r0 + tid] = RG;
    atomicAdd(&scal[2], R2);   // Sum_ij G*relu^2 (for l_order)
  }
}

// ---------------------------------------------------------------------------
// Combine column segments, loss + tau gradient + done flag (single block).
// ---------------------------------------------------------------------------
__global__ void k_loss_grad(const float* Pp, const float* Zp, const float* Rp,
                            const float* Cg,
                            float* Pnz, float* Zc, float* Rg,
                            float* gtau, float* scal) {
  __shared__ float red[1024];
  float acc = 0.f;
  for (int i = threadIdx.x; i < NF; i += 1024) {
    float P = 1.f, Z = 0.f, RG = 0.f;
#pragma unroll
    for (int s = 0; s < CS; s++) {
      P  *= Pp[(size_t)s * NF + i];
      Z  += Zp[(size_t)s * NF + i];
      RG += Rp[(size_t)s * NF + i];
    }
    Pnz[i] = P; Zc[i] = Z; Rg[i] = RG;
    float Pf = (Z > 0.f) ? 0.f : P;
    acc += 0.25f * (Pf + 1.f) * (Pf + 1.f);
  }
  red[threadIdx.x] = acc;
  __syncthreads();
  for (int s = 512; s > 0; s >>= 1) {
    if (threadIdx.x < s) red[threadIdx.x] += red[threadIdx.x + s];
    __syncthreads();
  }
  if (threadIdx.x == 0) {
    float lodd   = red[0] / (float)NF;
    float lorder = scal[2] / ((float)NF * (float)NT);
    float loss   = lodd + 0.5f * lorder;
    scal[4] = loss;
    scal[1] = (scal[0] != 0.f || loss < TOLC) ? 1.f : 0.f;  // done_cur
  }
  const float inv = 1.f / ((float)NF * (float)NT);
  for (int k = threadIdx.x; k < NN; k += 1024) {
    float g = 0.f;
    if (k < NF)   g += Rg[k];
    if (k >= NIN) g -= Cg[k - NIN];
    gtau[k] = g * inv;
  }
}

// ---------------------------------------------------------------------------
// Pass 2: elementwise gradient + Adam on GL/m/v (float4 streams).
// One block per row. Uses pre-update tau. Gated on done_prev.
// ---------------------------------------------------------------------------
__global__ __launch_bounds__(256) void k_pass2(
    const float* __restrict__ A, float* GL, float* mG, float* vG,
    const float* __restrict__ tau, const float* Pnz, const float* Zc,
    const float* scal, float ibc1, float ibc2) {
  if (scal[0] != 0.f) return;  // done_prev: params frozen
  const int i = blockIdx.x;
  const float pnz = Pnz[i];
  const float zc  = Zc[i];
  const float Pfull = (zc > 0.f) ? 0.f : pnz;
  const float codd  = (0.5f / (float)NF) * (Pfull + 1.f);
  const float taui  = tau[i];
  const float cord  = 0.5f / ((float)NF * (float)NT);

  const float4* a4p  = (const float4*)(A + (size_t)i * APITCH + NIN);
  const float4* tv4p = (const float4*)(tau + NIN);
  float4* gl4p = (float4*)(GL + (size_t)i * NT);
  float4* m4p  = (float4*)(mG + (size_t)i * NT);
  float4* v4p  = (float4*)(vG + (size_t)i * NT);
  const int NT4 = NT / 4;   // 764

  for (int j4 = threadIdx.x; j4 < NT4; j4 += 256) {
    const float4 a4  = a4p[j4];
    const float4 gl4 = gl4p[j4];
    const float4 m4i = m4p[j4];
    const float4 v4i = v4p[j4];
    const float4 tv4 = tv4p[j4];
    float4 glo, mo, vo;
#pragma unroll
    for (int c = 0; c < 4; c++) {
      const float a  = (&a4.x)[c];
      const float gl = (&gl4.x)[c];
      const float g  = sigmoidf(gl);
      const float f  = 1.f - 2.f * a * g;
      const float excl = (zc == 0.f) ? (pnz * fast_rcp(f))
                        : ((zc == 1.f && f == 0.f) ? pnz : 0.f);
      float d = taui - (&tv4.x)[c] + OEPS; d = d > 0.f ? d : 0.f;
      const float gG  = codd * excl * (-2.f * a) + cord * d * d;
      const float gGL = gG * g * (1.f - g);
      const float m = BB1 * (&m4i.x)[c] + (1.f - BB1) * gGL;
      const float v = BB2 * (&v4i.x)[c] + (1.f - BB2) * gGL * gGL;
      (&mo.x)[c] = m;
      (&vo.x)[c] = v;
      (&glo.x)[c] = gl - LRC * (m * ibc1) * fast_rcp(sqrtf(v * ibc2) + AEPS);
    }
    m4p[j4]  = mo;
    v4p[j4]  = vo;
    gl4p[j4] = glo;
  }
}

// ---------------------------------------------------------------------------
// tau Adam update (after pass 2, so pass 2 saw old tau). Gated on done_prev.
// ---------------------------------------------------------------------------
__global__ void k_tau(float* tau, float* mt, float* vt, const float* gtau,
                      const float* scal, float ibc1, float ibc2) {
  int k = blockIdx.x * blockDim.x + threadIdx.x;
  if (k >= NN) return;
  if (scal[0] != 0.f) return;
  const float g = gtau[k];
  const float m = BB1 * mt[k] + (1.f - BB1) * g;
  const float v = BB2 * vt[k] + (1.f - BB2) * g * g;
  mt[k] = m; vt[k] = v;
  tau[k] -= LRC * (m * ibc1) * fast_rcp(sqrtf(v * ibc2) + AEPS);
}

// ---------------------------------------------------------------------------
// Final loss -> d_out[0] (k_loss_grad already computed scal[4])
// ---------------------------------------------------------------------------
__global__ void k_final(const float* scal, float* out) {
  if (threadIdx.x == 0 && blockIdx.x == 0) out[0] = scal[4];
}

// ---------------------------------------------------------------------------
extern "C" void kernel_launch(void* const* d_in, const int* in_sizes, int n_in,
                              void* d_out, int out_size, void* d_ws, size_t ws_size,
                              hipStream_t stream) {
  const float* A    = (const float*)d_in[0];   // (N, N)
  const float* tau0 = (const float*)d_in[1];   // (N,)
  const float* GL0  = (const float*)d_in[2];   // (NF, NT)

  float* ws = (float*)d_ws;
  const size_t MAT = (size_t)NF * NT;
  float* GL   = ws;
  float* mG   = ws + MAT;
  float* vG   = ws + 2 * MAT;
  float* tau  = ws + 3 * MAT;
  float* mt   = tau + NN;
  float* vt   = mt + NN;
  float* gtau = vt + NN;
  float* Pnz  = gtau + NN;
  float* Zc   = Pnz + NF;
  float* Rg   = Zc + NF;
  float* Cg   = Rg + NF;
  float* scal = Cg + NT;        // [0]=done_prev [1]=done_cur [2]=lorder_acc [4]=loss
  float* Pp   = scal + 8;       // [CS][NF] per-segment partial products
  float* Zp   = Pp + (size_t)CS * NF;
  float* Rp   = Zp + (size_t)CS * NF;

  k_init<<<4096, 256, 0, stream>>>(GL0, tau0, GL, mG, vG, tau, mt, vt, scal);

  dim3 g1(NTILES, CS);
  for (int t = 1; t <= ITERS; t++) {
    const float ibc1 = 1.f / (1.f - powf(BB1, (float)t));
    const float ibc2 = 1.f / (1.f - powf(BB2, (float)t));
    k_zero<<<12, 256, 0, stream>>>(Cg, scal);
    k_pass1<<<g1, 256, 0, stream>>>(A, GL, tau, Pp, Zp, Rp, Cg, scal);
    k_loss_grad<<<1, 1024, 0, stream>>>(Pp, Zp, Rp, Cg, Pnz, Zc, Rg, gtau, scal);
    k_pass2<<<NF, 256, 0, stream>>>(A, GL, mG, vG, tau, Pnz, Zc, scal, ibc1, ibc2);
    k_tau<<<12, 256, 0, stream>>>(tau, mt, vt, gtau, scal, ibc1, ibc2);
  }

  // Loss at final parameters
  k_zero<<<12, 256, 0, stream>>>(Cg, scal);
  k_pass1<<<g1, 256, 0, stream>>>(A, GL, tau, Pp, Zp, Rp, Cg, scal);
  k_loss_grad<<<1, 1024, 0, stream>>>(Pp, Zp, Rp, Cg, Pnz, Zc, Rg, gtau, scal);
  k_final<<<1, 64, 0, stream>>>(scal, (float*)d_out);

  (void)in_sizes; (void)n_in; (void)out_size; (void)ws_size;
}